// MasifLigandNet_10703058501841
// MI455X (gfx1250) — compile-verified
//
#include <hip/hip_runtime.h>
#include <hip/hip_bf16.h>
#include <math.h>

// ---------------------------------------------------------------------------
// MasifLigandNet on MI455X (gfx1250, wave32)
//   K1: per-(batch, 16-ligand-tile, vertex-chunk) wave computes score tiles
//       s[i][j] = |p_i|^2 - 2 p_i . l_j   via V_WMMA_F32_16X16X4_F32
//       (rank-equivalent to squared distance for fixed j). Per-lane top-10
//       kept in registers; a branch-free min-reduce over the 8 WMMA outputs
//       gates the (rare) insertion path so the steady-state loop is just
//       load_b96 + wmma + 7x v_min + 1 compare.
//   K2: merges per-chunk candidate lists -> final top-10 indices per (b,j).
//   K3: deterministic dedupe (first-occurrence, no atomics), mean-pool of
//       unique vertex features, then Linear -> BN(eval) -> SiLU -> Linear.
// ---------------------------------------------------------------------------

#define KNN 10

typedef __attribute__((ext_vector_type(2))) float v2f;
typedef __attribute__((ext_vector_type(8))) float v8f;

// Branch-free sorted insert: list ascending, best[KNN-1] is the cutoff.
// One compare per slot drives min/max (values) + 2 cndmask (indices).
__device__ __forceinline__ void topk_insert(float (&best)[KNN], int (&bidx)[KNN],
                                            float v, int vi)
{
    float cv = v; int ci = vi;
#pragma unroll
    for (int s = 0; s < KNN; ++s) {
        const float bs = best[s];
        const int   bi = bidx[s];
        const bool  lt = cv < bs;
        best[s] = lt ? cv : bs;   // min
        bidx[s] = lt ? ci : bi;
        cv      = lt ? bs : cv;   // max carries down
        ci      = lt ? bi : ci;
    }
}

__global__ __launch_bounds__(32)
void k1_scores_topk(const float* __restrict__ pos,     // [B,N,3]
                    const float* __restrict__ lig,     // [B,L,3]
                    float* __restrict__ candV,         // [B,L,nchunks,20]
                    int*   __restrict__ candI,         // [B,L,nchunks,20]
                    int B, int N, int L, int nchunks, int chunk)
{
    const int lt   = L >> 4;                 // ligand tiles of 16
    const int b    = blockIdx.x / lt;
    const int jt   = blockIdx.x % lt;
    const int ck   = blockIdx.y;
    const int lane = threadIdx.x;            // wave32
    const int n16  = lane & 15;
    const bool hi  = (lane >= 16);

    // --- B matrix (4 x 16, f32): column n = (-2lx, -2ly, -2lz, 1) ---
    // Layout: lanes 0-15 hold K=0,1 in the two VGPRs; lanes 16-31 hold K=2,3.
    const float* lp = lig + ((size_t)b * L + (size_t)(jt * 16 + n16)) * 3;
    const float lx = lp[0], ly = lp[1], lz = lp[2];
    v2f bm;
    bm.x = hi ? (-2.0f * lz) : (-2.0f * lx);
    bm.y = hi ? 1.0f         : (-2.0f * ly);

    float best[KNN]; int bidx[KNN];
#pragma unroll
    for (int s = 0; s < KNN; ++s) { best[s] = 3.0e38f; bidx[s] = 0; }

    const int    vbase0 = ck * chunk;
    const int    ntiles = chunk >> 4;
    const float* pb     = pos + (size_t)b * N * 3;

#pragma unroll 2
    for (int t = 0; t < ntiles; ++t) {
        const int vrow = vbase0 + t * 16 + n16;       // vertex this lane loads
        const float* pp = pb + (size_t)vrow * 3;
        __builtin_prefetch(pp + 48, 0, 0);            // next tile (speculative)
        const float px = pp[0], py = pp[1], pz = pp[2];
        const float p2 = px * px + py * py + pz * pz;

        // A matrix (16 x 4, f32): row m = (px, py, pz, |p|^2)
        v2f am;
        am.x = hi ? pz : px;
        am.y = hi ? p2 : py;

        v8f c = {};
        // D = A x B + 0  -> v_wmma_f32_16x16x4_f32
        c = __builtin_amdgcn_wmma_f32_16x16x4_f32(
                /*neg_a=*/false, am, /*neg_b=*/false, bm,
                /*c_mod=*/(short)0, c, /*reuse_a=*/false, /*reuse_b=*/false);

        // Tile-level pre-filter: branch-free min tree over the 8 scores;
        // only enter the insert path if this tile can beat the cutoff.
        const float m0 = fminf(c[0], c[1]);
        const float m1 = fminf(c[2], c[3]);
        const float m2 = fminf(c[4], c[5]);
        const float m3 = fminf(c[6], c[7]);
        const float mt = fminf(fminf(m0, m1), fminf(m2, m3));

        if (mt < best[KNN - 1]) {             // rare (~10 tiles of 128/lane)
            // C layout: VGPR r holds M = r (lanes 0-15) or M = 8+r
            // (lanes 16-31), N = lane & 15 -> all 8 scores share one lig atom.
            const int mofs = vbase0 + t * 16 + (hi ? 8 : 0);
#pragma unroll
            for (int r = 0; r < 8; ++r) {
                const float v  = c[r];
                const int   vi = mofs + r;
                if (v < best[KNN - 1])
                    topk_insert(best, bidx, v, vi);
            }
        }
    }

    // two half-lane candidate lists per (b, j, chunk): 20 entries
    const int j = jt * 16 + n16;
    const size_t base =
        ((((size_t)b * L + j) * nchunks) + ck) * (2 * KNN) + (hi ? KNN : 0);
#pragma unroll
    for (int s = 0; s < KNN; ++s) {
        candV[base + s] = best[s];
        candI[base + s] = bidx[s];
    }
}

__global__ __launch_bounds__(256)
void k2_merge(const float* __restrict__ candV,
              const int*   __restrict__ candI,
              int*         __restrict__ selIdx,   // [B,L,KNN]
              int B, int L, int nchunks)
{
    const int pair = blockIdx.x * blockDim.x + threadIdx.x;  // (b*L + j)
    if (pair >= B * L) return;
    const int M = nchunks * 2 * KNN;
    const float* cv = candV + (size_t)pair * M;
    const int*   ci = candI + (size_t)pair * M;

    float best[KNN]; int bidx[KNN];
#pragma unroll
    for (int s = 0; s < KNN; ++s) { best[s] = 3.0e38f; bidx[s] = 0; }

    for (int m = 0; m < M; ++m) {
        const float v  = cv[m];
        const int   vi = ci[m];
        if (v < best[KNN - 1])
            topk_insert(best, bidx, v, vi);
    }
#pragma unroll
    for (int s = 0; s < KNN; ++s) selIdx[(size_t)pair * KNN + s] = bidx[s];
}

__global__ __launch_bounds__(256)
void k3_pool_head(const float* __restrict__ x,      // [B,N,D]
                  const int*   __restrict__ selIdx, // [B,L,KNN]
                  const float* __restrict__ W1,     // [D,D]
                  const float* __restrict__ b1,
                  const float* __restrict__ gamma,
                  const float* __restrict__ beta,
                  const float* __restrict__ rmean,
                  const float* __restrict__ rvar,
                  const float* __restrict__ W2,     // [D,OUT]
                  const float* __restrict__ b2,
                  float* __restrict__ out,          // [B,OUT]
                  int N, int L, int D, int OUT)
{
    constexpr int MAXSEL = 64 * KNN;  // 640
    constexpr int MAXD   = 128;
    __shared__ int           selS[MAXSEL];
    __shared__ unsigned char own[MAXSEL];
    __shared__ float         emb[MAXD];
    __shared__ float         act[MAXD];

    const int b   = blockIdx.x;
    const int tid = threadIdx.x;
    const int S   = L * KNN;

    for (int s = tid; s < S; s += blockDim.x)
        selS[s] = selIdx[(size_t)b * S + s];
    __syncthreads();

    // first-occurrence dedupe: deterministic, order-independent of scheduling
    for (int s = tid; s < S; s += blockDim.x) {
        const int idx = selS[s];
        unsigned char f = 1;
        for (int t = 0; t < s; ++t)
            if (selS[t] == idx) { f = 0; break; }
        own[s] = f;
    }
    __syncthreads();

    // mean-pool unique vertex features; thread d handles channel d (coalesced)
    if (tid < D) {
        float a = 0.0f; int cnt = 0;
        for (int s = 0; s < S; ++s) {
            if (own[s]) {
                a += x[((size_t)b * N + (size_t)selS[s]) * D + tid];
                ++cnt;
            }
        }
        emb[tid] = a / (float)cnt;
    }
    __syncthreads();

    // Linear(D,D) + BatchNorm(eval) + SiLU
    if (tid < D) {
        float h = b1[tid];
        for (int k = 0; k < D; ++k)
            h += emb[k] * W1[(size_t)k * D + tid];
        h = (h - rmean[tid]) * (1.0f / sqrtf(rvar[tid] + 1e-5f)) * gamma[tid]
            + beta[tid];
        act[tid] = h / (1.0f + __expf(-h));   // SiLU
    }
    __syncthreads();

    // Linear(D, OUT)
    if (tid < OUT) {
        float o = b2[tid];
        for (int d = 0; d < D; ++d)
            o += act[d] * W2[(size_t)d * OUT + tid];
        out[(size_t)b * OUT + tid] = o;
    }
}

extern "C" void kernel_launch(void* const* d_in, const int* in_sizes, int n_in,
                              void* d_out, int out_size, void* d_ws, size_t ws_size,
                              hipStream_t stream) {
    (void)n_in; (void)ws_size;
    const float* pos   = (const float*)d_in[0];
    const float* x     = (const float*)d_in[1];
    const float* lig   = (const float*)d_in[2];
    const float* W1    = (const float*)d_in[3];
    const float* b1    = (const float*)d_in[4];
    const float* gamma = (const float*)d_in[5];
    const float* beta  = (const float*)d_in[6];
    const float* rmean = (const float*)d_in[7];
    const float* rvar  = (const float*)d_in[8];
    const float* W2    = (const float*)d_in[9];
    const float* b2    = (const float*)d_in[10];
    float* out = (float*)d_out;

    const int OUT = in_sizes[10];                 // 7
    const int D   = in_sizes[4];                  // 128
    const int B   = out_size / OUT;               // 32
    const int L   = in_sizes[2] / (3 * B);        // 64
    const int N   = (int)((long long)in_sizes[0] / (3LL * B)); // 32768

    const int NCHUNKS = 16;
    const int chunk   = N / NCHUNKS;              // 2048 (multiple of 16)

    // workspace layout: candV | candI | selIdx  (~5.3 MB total)
    const size_t CB = (size_t)B * L * NCHUNKS * (2 * KNN);
    float* candV  = (float*)d_ws;
    int*   candI  = (int*)((char*)d_ws + CB * sizeof(float));
    int*   selIdx = (int*)((char*)d_ws + CB * (sizeof(float) + sizeof(int)));

    dim3 g1(B * (L / 16), NCHUNKS);
    k1_scores_topk<<<g1, 32, 0, stream>>>(pos, lig, candV, candI,
                                          B, N, L, NCHUNKS, chunk);

    const int pairs = B * L;
    k2_merge<<<(pairs + 255) / 256, 256, 0, stream>>>(candV, candI, selIdx,
                                                      B, L, NCHUNKS);

    k3_pool_head<<<B, 256, 0, stream>>>(x, selIdx, W1, b1, gamma, beta,
                                        rmean, rvar, W2, b2, out,
                                        N, L, D, OUT);
}